// SpatialLightModulator_10084583211460
// MI455X (gfx1250) — compile-verified
//
#include <hip/hip_runtime.h>
#include <stdint.h>
#include <math.h>

typedef float f32x2 __attribute__((ext_vector_type(2)));
typedef float f32x4 __attribute__((ext_vector_type(4)));

#define TWO_PI_F 6.283185307179586f

__device__ __forceinline__ float wrap_phase01(float p) {
    // wrap into [0, 2pi], matching reference's floor-division formulation
    if (p < 0.0f)
        p = p + TWO_PI_F * (1.0f + floorf((0.0f - p) / TWO_PI_F));
    else if (p > TWO_PI_F)
        p = p - TWO_PI_F * (1.0f + floorf((p - TWO_PI_F) / TWO_PI_F));
    return p;
}

__global__ __launch_bounds__(256) void slm_kernel(
    const float* __restrict__ u_real,
    const float* __restrict__ u_imag,
    const float* __restrict__ phase,
    const float* __restrict__ spectrum,
    float* __restrict__ out)
{
    const int W  = 2048, C = 4;
    const int Hs = 1080, Ws = 1920;

    __shared__ float sh[1936];   // 1920 phase floats + zero slot at [1920]

    const int tid  = threadIdx.x;
    const int blk  = blockIdx.x;
    const int row  = blk >> 1;     // output row 0..2047
    const int half = blk & 1;      // which half of the row

    // per-channel scale factors: lambda0 / lambda_c (uniform across block)
    const float s0  = spectrum[0];
    const float sc1 = s0 / spectrum[1];
    const float sc2 = s0 / spectrum[2];
    const float sc3 = s0 / spectrum[3];

    // nearest source row (round half-to-even == jnp.round)
    const float yscale = (float)(Hs - 1) / (float)(W - 1);   // 1079/2047
    const int   iy     = (int)rintf(fmaf((float)row, yscale, 0.5f));
    const bool  row_oob = (iy < 0) | (iy > Hs - 1);

    // ---- stage phase row into LDS via CDNA5 async data mover (ASYNCcnt) ----
    const unsigned lds_base = (unsigned)(uintptr_t)(&sh[0]); // low 32 bits of generic LDS addr = LDS byte offset
    if (!row_oob) {
        const float* prow = phase + (size_t)iy * Ws;
        #pragma unroll
        for (int k = 0; k < 2; ++k) {
            int idx = tid + k * 256;          // 480 x B128 covers 1920 floats
            if (idx < Ws / 4) {
                unsigned           lds_addr = lds_base + (unsigned)(idx * 16);
                unsigned long long gaddr    = (unsigned long long)(uintptr_t)(prow + idx * 4);
                asm volatile("global_load_async_to_lds_b128 %0, %1, off"
                             :: "v"(lds_addr), "v"(gaddr) : "memory");
            }
        }
    }
    if (tid == 0) sh[Ws] = 0.0f;              // zero slot for OOB column
    asm volatile("s_wait_asynccnt 0" ::: "memory");
    __syncthreads();

    const float xscale = (float)(Ws - 1) / (float)(W - 1);   // 1919/2047

    // unit = (pixel x, channel-pair); consecutive lanes -> contiguous 16B stores
    #pragma unroll
    for (int k = 0; k < 8; ++k) {
        int u  = half * 2048 + k * 256 + tid;  // unit index in [0, 4096)
        int x  = u >> 1;
        int c0 = (u & 1) * 2;                  // channels {0,1} or {2,3}

        int ix = (int)rintf(fmaf((float)x, xscale, 0.5f));
        float p;
        if (row_oob || ix < 0 || ix > Ws - 1) p = 0.0f;   // mode='constant', cval=0
        else                                  p = sh[ix];
        p = wrap_phase01(p);

        float pa = (c0 == 0) ? p       : p * sc2;
        float pb = (c0 == 0) ? p * sc1 : p * sc3;

        size_t base = ((size_t)row * W + (size_t)x) * C + (size_t)c0;
        f32x2 ur = __builtin_nontemporal_load((const f32x2*)(u_real + base));
        f32x2 ui = __builtin_nontemporal_load((const f32x2*)(u_imag + base));

        float sa, ca, sb, cb;
        __sincosf(pa, &sa, &ca);
        __sincosf(pb, &sb, &cb);

        f32x4 o;
        o.x = ur.x * ca - ui.x * sa;   // re, channel c0
        o.y = ur.x * sa + ui.x * ca;   // im, channel c0
        o.z = ur.y * cb - ui.y * sb;   // re, channel c0+1
        o.w = ur.y * sb + ui.y * cb;   // im, channel c0+1

        __builtin_nontemporal_store(o, (f32x4*)(out + base * 2));
    }
}

extern "C" void kernel_launch(void* const* d_in, const int* in_sizes, int n_in,
                              void* d_out, int out_size, void* d_ws, size_t ws_size,
                              hipStream_t stream) {
    const float* u_real   = (const float*)d_in[0];
    const float* u_imag   = (const float*)d_in[1];
    const float* phase    = (const float*)d_in[2];
    const float* spectrum = (const float*)d_in[3];
    float*       out      = (float*)d_out;

    dim3 grid(4096);   // 2048 rows x 2 half-rows
    dim3 block(256);   // 8 wave32s
    hipLaunchKernelGGL(slm_kernel, grid, block, 0, stream,
                       u_real, u_imag, phase, spectrum, out);
}